// LuongAttnDecoderRNN_17523466568183
// MI455X (gfx1250) — compile-verified
//
#include <hip/hip_runtime.h>
#include <cmath>

typedef __attribute__((ext_vector_type(2))) float v2f;
typedef __attribute__((ext_vector_type(8))) float v8f;

#define BATCH 64
#define SEQ   2048
#define HID   512
#define VOC   50257

// ---------------------------------------------------------------------------
// Generic WMMA f32 GEMM, M fixed = 64:
//   C[m][n] = act( sum_k A[m][k] * Bw[n][k] + bias[n] )
// A: [64][K] row-major, Bw: [N][K] row-major (computes A @ Bw^T).
// One block = 4 waves; block covers 64 N-columns (wave w owns columns
// [w*16, w*16+16)) and all 4 M-tiles. BOTH operands are staged through LDS in
// 64-wide K chunks with fully-coalesced float4 (b128) global loads, so the
// dominant out_W stream (103 MB) moves at clean 64B-line HBM efficiency.
// LDS pitch = 68 floats: keeps float4 staging stores 16B-aligned and makes
// the per-lane WMMA fragment reads (banks 4*lm + klo) conflict-free across
// all 32 lanes.
// ---------------------------------------------------------------------------
__global__ __launch_bounds__(128) void gemm64_wmma(
    const float* __restrict__ A, const float* __restrict__ Bw,
    const float* __restrict__ bias, float* __restrict__ C,
    int K, int N, int ldc, int act)
{
  constexpr int KC = 64;
  constexpr int LP = KC + 4;                  // LDS pitch (floats)
  __shared__ __align__(16) float alds[64 * LP];   // ~17 KB
  __shared__ __align__(16) float blds[64 * LP];   // ~17 KB

  const int tid   = threadIdx.x;
  const int wave  = tid >> 5;
  const int lane  = tid & 31;
  const int lm    = lane & 15;                // column (B) / row (A) in tile
  const int khalf = (lane >> 4) * 2;          // K pair selected by lane half
  const int nbase = blockIdx.x * 64;
  const int ncol  = nbase + wave * 16 + lm;

  v8f acc[4] = {};                            // 4 M-tiles of 16x16 f32

  for (int kb = 0; kb < K; kb += KC) {
    __syncthreads();
    // Cooperative float4-coalesced stage of A[0:64][kb:+KC] and the block's
    // 64 rows of Bw[nbase:+64][kb:+KC] into LDS.
    #pragma unroll
    for (int i = tid; i < 64 * (KC / 4); i += 128) {   // 1024 float4s, 8 iters
      const int r  = i / (KC / 4);
      const int c4 = (i % (KC / 4)) * 4;
      *(float4*)&alds[r * LP + c4] =
          *(const float4*)(A + (size_t)r * K + kb + c4);
      int br = nbase + r;
      br = (br < N) ? br : (N - 1);                    // clamp OOB tail rows
      const float* bptr = Bw + (size_t)br * K + kb + c4;
      *(float4*)&blds[r * LP + c4] = *(const float4*)bptr;
      if (kb + KC < K) __builtin_prefetch(bptr + KC, 0, 0); // global_prefetch
    }
    __syncthreads();

    const int brow = (wave * 16 + lm) * LP;
    #pragma unroll 4
    for (int k = 0; k < KC; k += 4) {
      const int klo = k + khalf;
      v2f bv;                                  // B[k][n] fragment from LDS
      bv.x = blds[brow + klo];
      bv.y = blds[brow + klo + 1];
      #pragma unroll
      for (int mt = 0; mt < 4; ++mt) {
        const int abase = (mt * 16 + lm) * LP + klo;
        v2f av;
        av.x = alds[abase];
        av.y = alds[abase + 1];
        acc[mt] = __builtin_amdgcn_wmma_f32_16x16x4_f32(
            false, av, false, bv, (short)0, acc[mt], false, false);
      }
    }
  }

  if (ncol < N) {
    const float bs = bias[ncol];
    #pragma unroll
    for (int mt = 0; mt < 4; ++mt) {
      #pragma unroll
      for (int j = 0; j < 8; ++j) {
        const int m = mt * 16 + j + 8 * (lane >> 4);   // C VGPR j -> rows j, j+8
        float v = acc[mt][j] + bs;
        if (act == 1) v = tanhf(v);
        C[(size_t)m * ldc + ncol] = v;                 // lanes 0-15 coalesced
      }
    }
  }
}

// ---------------------------------------------------------------------------
// Fused Luong-dot attention, one block per batch row, 8 waves.
// Single pass over encoder_outputs[b] (4 MB) with flash-style online softmax:
// per-wave running (max, sum, ctx[512]) in registers; raw energies stashed in
// LDS so attn_weights can be normalized without re-reading HBM.
// ---------------------------------------------------------------------------
__global__ __launch_bounds__(256) void attn_fused(
    const float* __restrict__ hnew, const float* __restrict__ enc,
    float* __restrict__ ctx_out, float* __restrict__ attn_out)
{
  __shared__ float hv[HID];
  __shared__ float energy[SEQ];
  __shared__ float ctxbuf[8 * HID];
  __shared__ float wm[8], wl[8];

  const int b    = blockIdx.x;
  const int tid  = threadIdx.x;
  const int wave = tid >> 5;
  const int lane = tid & 31;

  for (int i = tid; i < HID; i += 256) hv[i] = hnew[b * HID + i];
  __syncthreads();

  const float4* encb = (const float4*)(enc + (size_t)b * SEQ * HID);
  const float4* hv4  = (const float4*)hv;

  float m = -INFINITY, l = 0.0f;
  float4 cacc[4];
  #pragma unroll
  for (int j = 0; j < 4; ++j) cacc[j] = make_float4(0.f, 0.f, 0.f, 0.f);

  for (int s = wave * 256; s < (wave + 1) * 256; ++s) {
    const float4* row = encb + (size_t)s * (HID / 4);
    float4 ev[4];
    float dot = 0.0f;
    #pragma unroll
    for (int j = 0; j < 4; ++j) {
      ev[j] = row[lane + 32 * j];                // coalesced b128 loads
      const float4 hq = hv4[lane + 32 * j];
      dot += ev[j].x * hq.x + ev[j].y * hq.y + ev[j].z * hq.z + ev[j].w * hq.w;
    }
    #pragma unroll
    for (int off = 16; off > 0; off >>= 1) dot += __shfl_xor(dot, off, 32);
    if (lane == 0) energy[s] = dot;

    const float mnew = fmaxf(m, dot);
    const float sc   = __expf(m - mnew);         // first iter: exp(-inf)=0
    const float p    = __expf(dot - mnew);
    l = l * sc + p;
    #pragma unroll
    for (int j = 0; j < 4; ++j) {
      cacc[j].x = cacc[j].x * sc + p * ev[j].x;
      cacc[j].y = cacc[j].y * sc + p * ev[j].y;
      cacc[j].z = cacc[j].z * sc + p * ev[j].z;
      cacc[j].w = cacc[j].w * sc + p * ev[j].w;
    }
    m = mnew;
  }

  if (lane == 0) { wm[wave] = m; wl[wave] = l; }
  __syncthreads();

  float M = -INFINITY;
  #pragma unroll
  for (int w = 0; w < 8; ++w) M = fmaxf(M, wm[w]);
  float L = 0.0f;
  #pragma unroll
  for (int w = 0; w < 8; ++w) L += wl[w] * __expf(wm[w] - M);

  const float ms = __expf(m - M);
  float4* cb4 = (float4*)&ctxbuf[wave * HID];
  #pragma unroll
  for (int j = 0; j < 4; ++j) {
    float4 v;
    v.x = cacc[j].x * ms; v.y = cacc[j].y * ms;
    v.z = cacc[j].z * ms; v.w = cacc[j].w * ms;
    cb4[lane + 32 * j] = v;
  }
  __syncthreads();

  const float invL = 1.0f / L;
  for (int c = tid; c < HID; c += 256) {
    float s = 0.0f;
    #pragma unroll
    for (int w = 0; w < 8; ++w) s += ctxbuf[w * HID + c];
    ctx_out[b * HID + c] = s * invL;
  }
  for (int s2 = tid; s2 < SEQ; s2 += 256) {
    attn_out[(size_t)b * SEQ + s2] = __expf(energy[s2] - M) * invL;
  }
}

// ---------------------------------------------------------------------------
// Elementwise helpers
// ---------------------------------------------------------------------------
__global__ void gather_concat(const int* __restrict__ seq,
                              const float* __restrict__ emb,
                              const float* __restrict__ aux,
                              float* __restrict__ xin)  // [64][1024]
{
  int idx = blockIdx.x * blockDim.x + threadIdx.x;
  if (idx >= BATCH * 2 * HID) return;
  int mrow = idx >> 10;
  int c    = idx & 1023;
  xin[idx] = (c < HID) ? emb[(size_t)seq[mrow] * HID + c]
                       : aux[mrow * HID + (c - HID)];
}

__device__ __forceinline__ float sigmoidf(float x) {
  return 1.0f / (1.0f + __expf(-x));
}

__global__ void gru_elem(const float* __restrict__ gx,
                         const float* __restrict__ gh,
                         const float* __restrict__ hprev,
                         float* __restrict__ hnew,
                         float* __restrict__ hnew_out)
{
  int idx = blockIdx.x * blockDim.x + threadIdx.x;
  if (idx >= BATCH * HID) return;
  int mrow = idx >> 9;
  int c    = idx & (HID - 1);
  const int base = mrow * 3 * HID;
  float r  = sigmoidf(gx[base + c]           + gh[base + c]);
  float z  = sigmoidf(gx[base + HID + c]     + gh[base + HID + c]);
  float ng = tanhf   (gx[base + 2 * HID + c] + r * gh[base + 2 * HID + c]);
  float h  = hprev[idx];
  float hn = (1.0f - z) * ng + z * h;
  hnew[idx]     = hn;
  hnew_out[idx] = hn;
}

__global__ void concat2(const float* __restrict__ a,
                        const float* __restrict__ bm,
                        float* __restrict__ cat)  // [64][1024] = [a | bm]
{
  int idx = blockIdx.x * blockDim.x + threadIdx.x;
  if (idx >= BATCH * 2 * HID) return;
  int mrow = idx >> 10;
  int c    = idx & 1023;
  cat[idx] = (c < HID) ? a[mrow * HID + c] : bm[mrow * HID + (c - HID)];
}

// ---------------------------------------------------------------------------
extern "C" void kernel_launch(void* const* d_in, const int* in_sizes, int n_in,
                              void* d_out, int out_size, void* d_ws, size_t ws_size,
                              hipStream_t stream)
{
  const int*   seq   = (const int*)  d_in[0];
  const float* aux   = (const float*)d_in[1];
  const float* hprev = (const float*)d_in[2];   // [1,64,512]
  const float* enc   = (const float*)d_in[3];
  const float* emb   = (const float*)d_in[4];
  const float* cW    = (const float*)d_in[5];
  const float* cb    = (const float*)d_in[6];
  const float* Wih   = (const float*)d_in[7];
  const float* Whh   = (const float*)d_in[8];
  const float* bih   = (const float*)d_in[9];
  const float* bhh   = (const float*)d_in[10];
  const float* ccW   = (const float*)d_in[11];
  const float* ccb   = (const float*)d_in[12];
  const float* oW    = (const float*)d_in[13];
  const float* ob    = (const float*)d_in[14];

  float* out   = (float*)d_out;                       // [64][V]
  float* out_h = out + (size_t)BATCH * VOC;           // [1][64][512]
  float* out_a = out_h + BATCH * HID;                 // [64][1][2048]

  float* ws    = (float*)d_ws;
  float* xin   = ws;  ws += BATCH * 2 * HID;          // [64][1024]
  float* xcomb = ws;  ws += BATCH * HID;              // [64][512]
  float* gx    = ws;  ws += BATCH * 3 * HID;          // [64][1536]
  float* gh    = ws;  ws += BATCH * 3 * HID;          // [64][1536]
  float* hnew  = ws;  ws += BATCH * HID;              // [64][512]
  float* ctx   = ws;  ws += BATCH * HID;              // [64][512]
  float* cat2  = ws;  ws += BATCH * 2 * HID;          // [64][1024]
  float* cco   = ws;  ws += BATCH * HID;              // [64][512]

  // 1) x_in = [emb[seq] | aux]
  gather_concat<<<(BATCH * 2 * HID + 255) / 256, 256, 0, stream>>>(seq, emb, aux, xin);
  // 2) x = x_in @ combine_W^T + combine_b
  gemm64_wmma<<<dim3((HID + 63) / 64), 128, 0, stream>>>(xin, cW, cb, xcomb,
                                                         2 * HID, HID, HID, 0);
  // 3) gx = x @ W_ih^T + b_ih ; gh = h @ W_hh^T + b_hh
  gemm64_wmma<<<dim3((3 * HID + 63) / 64), 128, 0, stream>>>(xcomb, Wih, bih, gx,
                                                             HID, 3 * HID, 3 * HID, 0);
  gemm64_wmma<<<dim3((3 * HID + 63) / 64), 128, 0, stream>>>(hprev, Whh, bhh, gh,
                                                             HID, 3 * HID, 3 * HID, 0);
  // 4) GRU nonlinearity -> h_new (ws + output slot 2)
  gru_elem<<<(BATCH * HID + 255) / 256, 256, 0, stream>>>(gx, gh, hprev, hnew, out_h);
  // 5) fused attention: energies + softmax + context in ONE pass over enc
  attn_fused<<<BATCH, 256, 0, stream>>>(hnew, enc, ctx, out_a);
  // 6) concat_output = tanh([h_new | ctx] @ concat_W^T + concat_b)
  concat2<<<(BATCH * 2 * HID + 255) / 256, 256, 0, stream>>>(hnew, ctx, cat2);
  gemm64_wmma<<<dim3((HID + 63) / 64), 128, 0, stream>>>(cat2, ccW, ccb, cco,
                                                         2 * HID, HID, HID, 1);
  // 7) logits = concat_output @ out_W^T + out_b  (roofline term: 103MB of oW)
  gemm64_wmma<<<dim3((VOC + 63) / 64), 128, 0, stream>>>(cco, oW, ob, out,
                                                         HID, VOC, VOC, 0);
}